// CostVolume_42932493091434
// MI455X (gfx1250) — compile-verified
//
#include <hip/hip_runtime.h>

typedef __attribute__((ext_vector_type(2))) float v2f;
typedef __attribute__((ext_vector_type(8))) float v8f;

#define SD   48    // D == H == W
#define SC   64    // channels
#define NOFF 125   // 5^3 offsets
#define XR   52    // padded x-rows per LDS slot (48 + 2 halo each side)
#define RS   68    // row stride in floats (68 -> conflict-free ds_load_b64)
#define SLOTF (XR * RS)          // floats per wave slot (3536)

// One workgroup per (b, z, y) row. 4 waves share the 25 (dz,dy) combos.
// Per combo, a wave async-copies the shifted warp row into its private LDS
// slot (halo columns pre-zeroed => zero padding with no control flow), then
// runs the banded-Gram fp32 WMMA (16x16x4, K=64 over 16 steps) against c1
// A-fragments held in registers, extracts the +/-2 band, applies mean+leaky
// ReLU, and stores.
__global__ __launch_bounds__(128) void costvol_wmma_kernel(
    const float* __restrict__ c1, const float* __restrict__ warp,
    float* __restrict__ out) {
  __shared__ float smem[4][SLOTF];   // 56576 B total

  const int y    = blockIdx.x;
  const int z    = blockIdx.y;
  const int b    = blockIdx.z;
  const int lane = threadIdx.x & 31;
  const int wave = threadIdx.x >> 5;

  const int jn = lane & 15;   // N column / M row within lane group
  const int hi = lane >> 4;   // lane group 0/1
  const int kb = hi * 2;      // channel sub-offset within a K=4 step

  float* slot = &smem[wave][0];

  // ---- zero the halo rows (x = -2,-1,48,49 ; channels 0..63) once ----
  {
    const v2f z2 = {0.f, 0.f};
    const int c2 = lane * 2;                 // 0..62
    *(v2f*)&slot[0 * RS + c2]  = z2;
    *(v2f*)&slot[1 * RS + c2]  = z2;
    *(v2f*)&slot[50 * RS + c2] = z2;
    *(v2f*)&slot[51 * RS + c2] = z2;
  }

  const int rowvox = ((b * SD + z) * SD + y) * SD;   // row base (voxels)
  const int outrow = rowvox * NOFF;

  // ---- hoist all A fragments (c1 row) into registers: 3 tiles x 16 ksteps ----
  v2f afrag[3][16];
  {
    const float* c1row = c1 + rowvox * SC;
#pragma unroll
    for (int mi = 0; mi < 3; ++mi) {
      const float* ap = c1row + (mi * 16 + jn) * SC + kb;
#pragma unroll
      for (int k = 0; k < 16; ++k) afrag[mi][k] = *(const v2f*)(ap + 4 * k);
    }
  }

  // LDS byte address of this wave's slot (low 32 bits of generic address).
  const unsigned slot_lds = (unsigned)(unsigned long long)slot;

  for (int combo = wave; combo < 25; combo += 4) {
    const int dz = combo / 5;
    const int dy = combo % 5;
    const int zw = z + dz - 2;
    const int yw = y + dy - 2;
    const bool rowvalid = (zw >= 0) && (zw < SD) && (yw >= 0) && (yw < SD);

    if (rowvalid) {
      // prior ds reads of this slot must be done before overwriting it
      asm volatile("s_wait_dscnt 0x0" ::: "memory");
      // async-copy warp row (48 x 64 fp32 = 12KB) into slot rows 2..49.
      // iteration i: lanes 0-15 -> x = 2i (ch jn*4..+3), lanes 16-31 -> x = 2i+1
      const float* wrow = warp + ((b * SD + zw) * SD + yw) * (SD * SC);
      unsigned long long g = (unsigned long long)wrow + (unsigned)(lane * 16);
      unsigned l = slot_lds + (unsigned)((hi + 2) * (RS * 4) + jn * 16);
#pragma unroll
      for (int i = 0; i < 24; ++i) {
        asm volatile("global_load_async_to_lds_b128 %0, %1, off"
                     :: "v"(l), "v"(g) : "memory");
        g += (unsigned)(2 * SC * 4);   // 2 x-positions in global
        l += (unsigned)(2 * RS * 4);   // 2 rows in LDS
      }
      asm volatile("s_wait_asynccnt 0x0" ::: "memory");
    }

    const int offb = combo * 5;
    for (int mi = 0; mi < 3; ++mi) {
      const int x0 = mi * 16;
      v8f acc0 = {0.f, 0.f, 0.f, 0.f, 0.f, 0.f, 0.f, 0.f};
      v8f acc1 = {0.f, 0.f, 0.f, 0.f, 0.f, 0.f, 0.f, 0.f};

      if (rowvalid) {
        // B tile0 columns xw = x0-2+jn  -> slot row x0+jn     (halo covers xw<0)
        // B tile1 columns xw = x0+14+jn -> slot row x0+16+jn  (halo covers xw>47;
        //   rows past the slot feed only D columns that are never extracted)
        const float* bp0 = slot + (x0 + jn) * RS + kb;
        const float* bp1 = slot + (x0 + 16 + jn) * RS + kb;
#pragma unroll
        for (int k = 0; k < 16; ++k) {
          const v2f b0 = *(const v2f*)(bp0 + 4 * k);
          const v2f b1 = *(const v2f*)(bp1 + 4 * k);
          acc0 = __builtin_amdgcn_wmma_f32_16x16x4_f32(
              false, afrag[mi][k], false, b0, (short)0, acc0, false, false);
          acc1 = __builtin_amdgcn_wmma_f32_16x16x4_f32(
              false, afrag[mi][k], false, b1, (short)0, acc1, false, false);
        }
      }

      // ---- band extraction: D element (m', j') -> lane (j' + 16*(m'>=8)),
      //      VGPR m'%8. Accs are zero for invalid rows => leaky(0)=0. ----
#pragma unroll
      for (int v = 0; v < 8; ++v) {
        const int mp = v + 8 * hi;
        const int x  = x0 + mp;
        const int dx0 = jn - 2 - mp;       // tile0: xw = x0-2+jn
        if (dx0 >= -2 && dx0 <= 2) {
          float val = acc0[v] * (1.0f / 64.0f);
          val = fmaxf(val, 0.1f * val);    // leaky ReLU slope 0.1
          out[outrow + x * NOFF + offb + (dx0 + 2)] = val;
        }
        const int dx1 = jn + 14 - mp;      // tile1: xw = x0+14+jn
        if (dx1 >= -2 && dx1 <= 2) {
          float val = acc1[v] * (1.0f / 64.0f);
          val = fmaxf(val, 0.1f * val);
          out[outrow + x * NOFF + offb + (dx1 + 2)] = val;
        }
      }
    }
  }
}

extern "C" void kernel_launch(void* const* d_in, const int* in_sizes, int n_in,
                              void* d_out, int out_size, void* d_ws,
                              size_t ws_size, hipStream_t stream) {
  (void)in_sizes; (void)n_in; (void)out_size; (void)d_ws; (void)ws_size;
  const float* c1   = (const float*)d_in[0];
  const float* warp = (const float*)d_in[1];
  float* out        = (float*)d_out;

  dim3 grid(SD, SD, 2);   // (y, z, b) -> 4608 workgroups
  costvol_wmma_kernel<<<grid, 128, 0, stream>>>(c1, warp, out);
}